// Stage2RenderLoss_15771119911334
// MI455X (gfx1250) — compile-verified
//
#include <hip/hip_runtime.h>
#include <hip/hip_bf16.h>

// ---------------------------------------------------------------------------
// Voxel segment-mean merge of Gaussian attributes (N=2M points, K=4 colors).
// Memory/atomic bound. gfx1250 paths: double-buffered tensor_load_to_lds +
// s_wait_tensorcnt (TDM DMA pipeline), global_atomic_add_f32 scatter-reduce,
// global_prefetch_b8, ds scans/reductions. No WMMA: no matrix contraction here.
// ---------------------------------------------------------------------------

#define THREADS      256
#define GS_BLOCKS    2048
#define KEY_BLOCKS   256
#define SCAN_BLOCKS  1024
#define ACC_BLOCKS   2048
#define TDM_PTS      256          // points staged per wave per TDM op (768 dwords)

#define INT_MAX_C  2147483647
#define INT_MIN_C  (-2147483647 - 1)

typedef unsigned int uint32x4 __attribute__((ext_vector_type(4)));
typedef int          int32x4  __attribute__((ext_vector_type(4)));
typedef int          int32x8  __attribute__((ext_vector_type(8)));

#if __has_builtin(__builtin_amdgcn_tensor_load_to_lds)
#define HAS_TDM 1
#else
#define HAS_TDM 0
#endif

__device__ __forceinline__ int vfloor001(float x) {
    return (int)floorf(x / 0.01f);           // matches jnp.floor(means / 0.01) in f32
}

#if HAS_TDM
__device__ __forceinline__ unsigned lds_off(const void* p) {
    return (unsigned)(unsigned long long)(__attribute__((address_space(3))) const void*)p;
}

// 1-D contiguous f32 tile DMA: global -> LDS, per ISA D# layout (CDNA5 ISA)
__device__ __forceinline__ void tdm_load_f32_1d(unsigned lds_byte, const float* gsrc,
                                                unsigned tile_elems, unsigned remaining_elems) {
    unsigned long long ga = (unsigned long long)(const void*)gsrc;
    uint32x4 g0;
    g0.x = 1u;                                                  // count=1, user descriptor
    g0.y = lds_byte;                                            // lds_addr (bytes)
    g0.z = (unsigned)(ga & 0xffffffffull);                      // global_addr[31:0]
    g0.w = (unsigned)((ga >> 32) & 0x01ffffffull) | (2u << 30); // global_addr[56:32] | type=2
    int32x8 g1;
    g1[0] = (int)(2u << 16);                                    // data_size=2 -> 4 bytes
    g1[1] = (int)((remaining_elems & 0xffffu) << 16);           // tensor_dim0[15:0] @ bits63:48
    g1[2] = (int)(((remaining_elems >> 16) & 0xffffu) | (1u << 16)); // dim0[31:16] | tensor_dim1=1
    g1[3] = (int)((tile_elems & 0xffffu) << 16);                // tile_dim0 @ bits127:112
    g1[4] = 0;                                                  // tile_dim1=0, tile_dim2=0 (unused)
    g1[5] = (int)remaining_elems;                               // tensor_dim0_stride[31:0]
    g1[6] = 0;
    g1[7] = 0;
    int32x4 z4 = {0, 0, 0, 0};
#if defined(__clang_major__) && (__clang_major__ >= 23)
    int32x8 z8 = {0, 0, 0, 0, 0, 0, 0, 0};
    __builtin_amdgcn_tensor_load_to_lds(g0, g1, z4, z4, z8, 0);
#else
    __builtin_amdgcn_tensor_load_to_lds(g0, g1, z4, z4, 0);
#endif
}

__device__ __forceinline__ void issue_chunk(const float* means, long long base, long long N,
                                            unsigned lds_byte) {
    unsigned remaining = (unsigned)((N - base) * 3);
    unsigned tile = remaining < (unsigned)(TDM_PTS * 3) ? remaining : (unsigned)(TDM_PTS * 3);
    asm volatile("" ::: "memory");
    tdm_load_f32_1d(lds_byte, means + base * 3, tile, remaining);
}
#endif

__device__ __forceinline__ void wait_tensorcnt_le0() {
#if __has_builtin(__builtin_amdgcn_s_wait_tensorcnt)
    __builtin_amdgcn_s_wait_tensorcnt((short)0);
#else
    asm volatile("s_wait_tensorcnt 0x0" ::: "memory");
#endif
    asm volatile("" ::: "memory");   // compiler barrier: LDS was written behind its back
}
__device__ __forceinline__ void wait_tensorcnt_le1() {
#if __has_builtin(__builtin_amdgcn_s_wait_tensorcnt)
    __builtin_amdgcn_s_wait_tensorcnt((short)1);
#else
    asm volatile("s_wait_tensorcnt 0x1" ::: "memory");
#endif
    asm volatile("" ::: "memory");
}

// ---------------- K0: zero outputs/counts (B128 path), init header ---------
__global__ void k_init(float* __restrict__ out, long long out_n,
                       float* __restrict__ counts, int N, int* __restrict__ hdr) {
    long long stride = (long long)gridDim.x * blockDim.x;
    long long gid = (long long)blockIdx.x * blockDim.x + threadIdx.x;
    const float4 z4 = make_float4(0.f, 0.f, 0.f, 0.f);

    float4* out4 = (float4*)out;
    long long n4 = out_n >> 2;
    for (long long i = gid; i < n4; i += stride) out4[i] = z4;
    for (long long i = (n4 << 2) + gid; i < out_n; i += stride) out[i] = 0.0f;

    float4* cnt4 = (float4*)counts;
    long long c4 = (long long)N >> 2;
    for (long long i = gid; i < c4; i += stride) cnt4[i] = z4;
    for (long long i = (c4 << 2) + gid; i < N; i += stride) counts[i] = 0.0f;

    if (blockIdx.x == 0 && threadIdx.x < 9) {
        int t = threadIdx.x;
        hdr[t] = (t < 3) ? INT_MAX_C : (t < 6 ? INT_MIN_C : 0);
    }
}

// ---------------- K1: per-axis min/max of voxel indices --------------------
__global__ void k_minmax(const float* __restrict__ means, int* __restrict__ hdr, int N) {
    __shared__ int sred[6][THREADS];
    int mn0 = INT_MAX_C, mn1 = INT_MAX_C, mn2 = INT_MAX_C;
    int mx0 = INT_MIN_C, mx1 = INT_MIN_C, mx2 = INT_MIN_C;
    long long stride = (long long)gridDim.x * blockDim.x;
    for (long long i = (long long)blockIdx.x * blockDim.x + threadIdx.x; i < N; i += stride) {
        int v0 = vfloor001(means[3 * i + 0]);
        int v1 = vfloor001(means[3 * i + 1]);
        int v2 = vfloor001(means[3 * i + 2]);
        mn0 = min(mn0, v0); mx0 = max(mx0, v0);
        mn1 = min(mn1, v1); mx1 = max(mx1, v1);
        mn2 = min(mn2, v2); mx2 = max(mx2, v2);
    }
    int t = threadIdx.x;
    sred[0][t] = mn0; sred[1][t] = mn1; sred[2][t] = mn2;
    sred[3][t] = mx0; sred[4][t] = mx1; sred[5][t] = mx2;
    __syncthreads();
    for (int off = THREADS / 2; off > 0; off >>= 1) {
        if (t < off) {
            sred[0][t] = min(sred[0][t], sred[0][t + off]);
            sred[1][t] = min(sred[1][t], sred[1][t + off]);
            sred[2][t] = min(sred[2][t], sred[2][t + off]);
            sred[3][t] = max(sred[3][t], sred[3][t + off]);
            sred[4][t] = max(sred[4][t], sred[4][t + off]);
            sred[5][t] = max(sred[5][t], sred[5][t + off]);
        }
        __syncthreads();
    }
    if (t == 0) {
        atomicMin(&hdr[0], sred[0][0]); atomicMin(&hdr[1], sred[1][0]); atomicMin(&hdr[2], sred[2][0]);
        atomicMax(&hdr[3], sred[3][0]); atomicMax(&hdr[4], sred[4][0]); atomicMax(&hdr[5], sred[5][0]);
    }
}

// ---------------- K2: grid dims + key range --------------------------------
__global__ void k_dims(int* __restrict__ hdr, long long cap) {
    int md0 = hdr[3] - hdr[0] + 1;
    int md1 = hdr[4] - hdr[1] + 1;
    int md2 = hdr[5] - hdr[2] + 1;
    int md12 = md1 * md2;                    // int32 arithmetic matches reference
    long long R = (long long)md0 * (long long)md12;
    if (R > cap) R = cap;
    if (R < 1)   R = 1;
    hdr[6] = md12;
    hdr[7] = md2;
    hdr[8] = (int)R;
}

// ---------------- K3: zero occupancy flags ---------------------------------
__global__ void k_zero_flags(int* __restrict__ flags, const int* __restrict__ hdr) {
    long long R = hdr[8];
    long long stride = (long long)gridDim.x * blockDim.x;
    for (long long i = (long long)blockIdx.x * blockDim.x + threadIdx.x; i < R; i += stride)
        flags[i] = 0;
}

// ---------------- K4: keys + occupancy (double-buffered TDM pipeline) ------
__global__ void k_keys(const float* __restrict__ means, const int* __restrict__ hdr,
                       int* __restrict__ keys, int* __restrict__ flags, int N) {
#if HAS_TDM
    __shared__ float stage[THREADS / 32][2][TDM_PTS * 3];   // 48 KB, 2 tiles per wave
#endif
    const int lane = threadIdx.x & 31;
    const int wib  = threadIdx.x >> 5;
    const int mn0 = hdr[0], mn1 = hdr[1], mn2 = hdr[2];
    const int md12 = hdr[6], md2 = hdr[7], R = hdr[8];

    const long long nchunks = ((long long)N + TDM_PTS - 1) / TDM_PTS;
    const long long nwaves  = (long long)gridDim.x * (blockDim.x >> 5);
    const long long wave0   = (long long)blockIdx.x * (blockDim.x >> 5) + wib;

#if HAS_TDM
    const unsigned lds_b0 = lds_off(&stage[wib][0][0]);
    const unsigned lds_b1 = lds_off(&stage[wib][1][0]);
    int buf = 0;
    if (wave0 < nchunks) issue_chunk(means, wave0 * TDM_PTS, N, lds_b0);   // prime pipeline
#endif
    for (long long c = wave0; c < nchunks; c += nwaves) {
        const long long base = c * TDM_PTS;
#if HAS_TDM
        const long long cn = c + nwaves;
        if (cn < nchunks) {
            issue_chunk(means, cn * TDM_PTS, N, buf ? lds_b0 : lds_b1);    // fill other buffer
            wait_tensorcnt_le1();                // in-order TDM: oldest (chunk c) complete
        } else {
            wait_tensorcnt_le0();
        }
#endif
#pragma unroll
        for (int r = 0; r < TDM_PTS / 32; ++r) {
            int p = r * 32 + lane;
            long long i = base + p;
            if (i < N) {
#if HAS_TDM
                float x = stage[wib][buf][3 * p + 0];
                float y = stage[wib][buf][3 * p + 1];
                float z = stage[wib][buf][3 * p + 2];
#else
                float x = means[3 * i + 0];
                float y = means[3 * i + 1];
                float z = means[3 * i + 2];
#endif
                int v0 = vfloor001(x) - mn0;
                int v1 = vfloor001(y) - mn1;
                int v2 = vfloor001(z) - mn2;
                int key = v0 * md12 + v1 * md2 + v2;
                keys[i] = key;
                if (key >= 0 && key < R) flags[key] = 1;
            }
        }
#if HAS_TDM
        buf ^= 1;
#endif
    }
}

// ---------------- K5a/b/c: exclusive prefix scan of flags ------------------
__global__ void k_scan_part(const int* __restrict__ flags, int* __restrict__ part,
                            const int* __restrict__ hdr) {
    __shared__ int s[THREADS];
    int R = hdr[8];
    int chunk = (R + SCAN_BLOCKS - 1) / SCAN_BLOCKS;
    int s0 = blockIdx.x * chunk;
    int s1 = min(s0 + chunk, R);
    int acc = 0;
    for (int i = s0 + threadIdx.x; i < s1; i += THREADS) acc += flags[i];
    s[threadIdx.x] = acc;
    __syncthreads();
    for (int off = THREADS / 2; off > 0; off >>= 1) {
        if (threadIdx.x < off) s[threadIdx.x] += s[threadIdx.x + off];
        __syncthreads();
    }
    if (threadIdx.x == 0) part[blockIdx.x] = s[0];
}

__global__ void k_scan_top(int* __restrict__ part) {   // 1 block of SCAN_BLOCKS threads
    __shared__ int s[SCAN_BLOCKS];
    int t = threadIdx.x;
    int x = part[t];
    s[t] = x;
    __syncthreads();
    for (int off = 1; off < SCAN_BLOCKS; off <<= 1) {
        int v = (t >= off) ? s[t - off] : 0;
        __syncthreads();
        s[t] += v;
        __syncthreads();
    }
    part[t] = s[t] - x;                                // exclusive
}

__global__ void k_scan_down(int* __restrict__ flags, const int* __restrict__ part,
                            const int* __restrict__ hdr) {
    __shared__ int s[THREADS];
    int R = hdr[8];
    int chunk = (R + SCAN_BLOCKS - 1) / SCAN_BLOCKS;
    int s0 = blockIdx.x * chunk;
    int s1 = min(s0 + chunk, R);
    int carry = part[blockIdx.x];
    int t = threadIdx.x;
    for (int base = s0; base < s1; base += THREADS) {
        int i = base + t;
        int x = (i < s1) ? flags[i] : 0;
        s[t] = x;
        __syncthreads();
        for (int off = 1; off < THREADS; off <<= 1) {
            int v = (t >= off) ? s[t - off] : 0;
            __syncthreads();
            s[t] += v;
            __syncthreads();
        }
        int incl  = s[t];
        int total = s[THREADS - 1];
        __syncthreads();
        if (i < s1) flags[i] = carry + (incl - x);     // exclusive prefix in-place
        carry += total;
    }
}

// ---------------- K6: scatter accumulate (f32 global atomics) --------------
__global__ void k_accum(const float* __restrict__ means, const float* __restrict__ scales,
                        const float* __restrict__ rots, const float* __restrict__ opac,
                        const float* __restrict__ colors, const int* __restrict__ keys,
                        const int* __restrict__ prefix, const int* __restrict__ hdr,
                        float* __restrict__ counts, float* __restrict__ out, int N, int K) {
    float* om   = out;
    float* os   = out + 3LL * N;
    float* orot = out + 6LL * N;
    float* oo   = out + 10LL * N;
    float* oc   = out + 11LL * N;
    const int R = hdr[8];
    const int CK = 3 * K;
    const bool vec4 = (CK & 3) == 0;
    const float4* rot4 = (const float4*)rots;
    const float4* col4 = (const float4*)colors;
    long long stride = (long long)gridDim.x * blockDim.x;
    for (long long i = (long long)blockIdx.x * blockDim.x + threadIdx.x; i < N; i += stride) {
        long long ip = i + stride;
        if (ip < N) {                                   // gfx1250 global_prefetch_b8
            __builtin_prefetch(&colors[ip * CK], 0, 0);
            __builtin_prefetch(&means[ip * 3], 0, 0);
        }
        int key = keys[i];
        int seg = (key >= 0 && key < R) ? prefix[key] : 0;
        long long s = seg;
        atomicAdd(&counts[s], 1.0f);
        atomicAdd(&om[3 * s + 0], means[3 * i + 0]);
        atomicAdd(&om[3 * s + 1], means[3 * i + 1]);
        atomicAdd(&om[3 * s + 2], means[3 * i + 2]);
        atomicAdd(&os[3 * s + 0], scales[3 * i + 0]);
        atomicAdd(&os[3 * s + 1], scales[3 * i + 1]);
        atomicAdd(&os[3 * s + 2], scales[3 * i + 2]);
        float4 rv = rot4[i];                            // rotations: 16B aligned
        atomicAdd(&orot[4 * s + 0], rv.x);
        atomicAdd(&orot[4 * s + 1], rv.y);
        atomicAdd(&orot[4 * s + 2], rv.z);
        atomicAdd(&orot[4 * s + 3], rv.w);
        atomicAdd(&oo[s], opac[i]);
        if (vec4) {                                     // colors: CK/4 x B128 loads
            long long b = i * (CK >> 2);
            for (int j = 0; j < (CK >> 2); ++j) {
                float4 cv = col4[b + j];
                atomicAdd(&oc[s * CK + 4 * j + 0], cv.x);
                atomicAdd(&oc[s * CK + 4 * j + 1], cv.y);
                atomicAdd(&oc[s * CK + 4 * j + 2], cv.z);
                atomicAdd(&oc[s * CK + 4 * j + 3], cv.w);
            }
        } else {
            for (int j = 0; j < CK; ++j)
                atomicAdd(&oc[s * CK + j], colors[i * (long long)CK + j]);
        }
    }
}

// ---------------- K7: finalize (divide by counts, normalize quats) ---------
__global__ void k_final(float* __restrict__ out, const float* __restrict__ counts, int N, int K) {
    float* om   = out;
    float* os   = out + 3LL * N;
    float* orot = out + 6LL * N;
    float* oo   = out + 10LL * N;
    float* oc   = out + 11LL * N;
    const int CK = 3 * K;
    const bool vec4 = (CK & 3) == 0;
    long long stride = (long long)gridDim.x * blockDim.x;
    for (long long s = (long long)blockIdx.x * blockDim.x + threadIdx.x; s < N; s += stride) {
        float c = counts[s];
        if (c > 0.0f) {
            om[3 * s + 0] /= c; om[3 * s + 1] /= c; om[3 * s + 2] /= c;
            os[3 * s + 0] /= c; os[3 * s + 1] /= c; os[3 * s + 2] /= c;
            oo[s] /= c;
            if (vec4) {
                float4* oc4 = (float4*)(oc + s * CK);
                for (int j = 0; j < (CK >> 2); ++j) {
                    float4 v = oc4[j];
                    v.x /= c; v.y /= c; v.z /= c; v.w /= c;
                    oc4[j] = v;
                }
            } else {
                for (int j = 0; j < CK; ++j) oc[s * CK + j] /= c;
            }
            float4* r4 = (float4*)(orot + 4 * s);
            float4 r = *r4;
            float nrm = sqrtf(r.x * r.x + r.y * r.y + r.z * r.z + r.w * r.w);
            float d = fmaxf(nrm, 1e-8f);
            r.x /= d; r.y /= d; r.z /= d; r.w /= d;
            *r4 = r;
        }
        // counts==0 rows were zeroed in k_init and never touched -> exact zeros
    }
}

// ---------------------------------------------------------------------------
extern "C" void kernel_launch(void* const* d_in, const int* in_sizes, int n_in,
                              void* d_out, int out_size, void* d_ws, size_t ws_size,
                              hipStream_t stream) {
    const float* means  = (const float*)d_in[0];
    const float* scales = (const float*)d_in[1];
    const float* rots   = (const float*)d_in[2];
    const float* opac   = (const float*)d_in[3];
    const float* colors = (const float*)d_in[4];
    float* out = (float*)d_out;

    const int N = in_sizes[3];                       // opacities: [N]
    const int K = in_sizes[4] / (3 * N);             // colors: [N,K,3] -> K=4

    // Workspace layout: hdr(64B) | keys(4N) | counts(4N) | partials(4*SCAN_BLOCKS) | flags(rest)
    char* ws = (char*)d_ws;
    int*   hdr      = (int*)ws;
    int*   keys     = (int*)(ws + 64);
    float* counts   = (float*)(ws + 64 + 4LL * N);
    int*   partials = (int*)(ws + 64 + 8LL * N);
    int*   flags    = (int*)(ws + 64 + 8LL * N + 4LL * SCAN_BLOCKS);
    long long fixed = 64 + 8LL * N + 4LL * SCAN_BLOCKS;
    long long cap   = ((long long)ws_size - fixed) / 4;
    if (cap < 1) cap = 1;

    k_init<<<GS_BLOCKS, THREADS, 0, stream>>>(out, (long long)out_size, counts, N, hdr);
    k_minmax<<<GS_BLOCKS, THREADS, 0, stream>>>(means, hdr, N);
    k_dims<<<1, 1, 0, stream>>>(hdr, cap);
    k_zero_flags<<<GS_BLOCKS, THREADS, 0, stream>>>(flags, hdr);
    k_keys<<<KEY_BLOCKS, THREADS, 0, stream>>>(means, hdr, keys, flags, N);
    k_scan_part<<<SCAN_BLOCKS, THREADS, 0, stream>>>(flags, partials, hdr);
    k_scan_top<<<1, SCAN_BLOCKS, 0, stream>>>(partials);
    k_scan_down<<<SCAN_BLOCKS, THREADS, 0, stream>>>(flags, partials, hdr);
    k_accum<<<ACC_BLOCKS, THREADS, 0, stream>>>(means, scales, rots, opac, colors,
                                                keys, flags, hdr, counts, out, N, K);
    k_final<<<ACC_BLOCKS, THREADS, 0, stream>>>(out, counts, N, K);
}